// CompGCNLayer_73942156968056
// MI455X (gfx1250) — compile-verified
//
#include <hip/hip_runtime.h>

#define N_NODES    100000
#define NUM_RELA   500
#define D          128
#define N_EDGES    1000000
#define HALF       (N_EDGES / 2)
#define KTOT       384                 // 3 * D concatenated K
#define LDS_STRIDE 388                 // 384 + 4 pad -> conflict-free banks
#define N_RELA_ROWS (2 * NUM_RELA + 1) // 1001
#define ROWS_PER_WG 32                 // two 16-row M tiles per wave, shared B

typedef float v2f __attribute__((ext_vector_type(2)));
typedef float v8f __attribute__((ext_vector_type(8)));

// ---------------- scratch init ----------------
__global__ void init_ws_kernel(int* __restrict__ p, int n) {
    int i = blockIdx.x * blockDim.x + threadIdx.x;
    if (i < n) p[i] = 0;
}

// ---------------- per-half last-edge index + in-degree count ----------------
__global__ void edge_scan_kernel(const int* __restrict__ edges,
                                 int* __restrict__ last0, int* __restrict__ cnt0,
                                 int* __restrict__ last1, int* __restrict__ cnt1) {
    int i = blockIdx.x * blockDim.x + threadIdx.x;
    if (i >= N_EDGES) return;
    int d = edges[i * 3 + 2];
    if (i < HALF) {
        atomicMax(&last0[d], i);
        atomicAdd(&cnt0[d], 1);
    } else {
        atomicMax(&last1[d], i - HALF);
        atomicAdd(&cnt1[d], 1);
    }
}

// ---------------- W_cat^T, K-pair interleaved: WtP[k/2][j][k&1] ----------------
__global__ void build_wt_kernel(const float* __restrict__ Wi,
                                const float* __restrict__ Wo,
                                const float* __restrict__ Ws,
                                float* __restrict__ WtP) {
    int idx = blockIdx.x * blockDim.x + threadIdx.x;
    if (idx >= KTOT * D) return;
    int k = idx >> 7;
    int j = idx & (D - 1);
    float v;
    if (k < D)          v = Wi[j * D + k];
    else if (k < 2 * D) v = Wo[j * D + (k - D)];
    else                v = Ws[j * D + (k - 2 * D)];
    WtP[(size_t)(k >> 1) * (2 * D) + j * 2 + (k & 1)] = v;
}

// ---------------- h_r = rela_embed @ W_r^T (tiny) ----------------
__global__ void hr_kernel(const float* __restrict__ rela,
                          const float* __restrict__ Wr,
                          float* __restrict__ out) {
    int idx = blockIdx.x * blockDim.x + threadIdx.x;
    if (idx >= N_RELA_ROWS * D) return;
    int i = idx >> 7;
    int j = idx & (D - 1);
    const float* a = rela + (size_t)i * D;
    const float* w = Wr + (size_t)j * D;
    float s = 0.f;
#pragma unroll 8
    for (int k = 0; k < D; ++k) s = fmaf(a[k], w[k], s);
    out[idx] = s;
}

// ---------------- main fused gather + GEMM (fp32 WMMA) ----------------
__launch_bounds__(256)
__global__ void hv_wmma_kernel(const float* __restrict__ node_embed,
                               const float* __restrict__ rela_embed,
                               const int* __restrict__ edges,
                               const int* __restrict__ last0,
                               const int* __restrict__ cnt0,
                               const int* __restrict__ last1,
                               const int* __restrict__ cnt1,
                               const float* __restrict__ WtP,
                               float* __restrict__ out) {
    __shared__ float ldsA[ROWS_PER_WG * LDS_STRIDE];

    const int rowBase = blockIdx.x * ROWS_PER_WG;
    const int t = threadIdx.x;

    // ---- stage A tile (32 rows x 384 K) into LDS: 16 threads per node row ----
#pragma unroll
    for (int rr = 0; rr < 2; ++rr) {
        const int r = (t >> 4) + rr * 16;  // node row 0..31
        const int l = t & 15;              // 16 lanes sweep K
        const int n = rowBase + r;
        const int c0 = cnt0[n];
        const int c1 = cnt1[n];
        int s0 = 0, rl0 = 0, s1 = 0, rl1 = 0;
        if (c0 > 0) { int e = last0[n];        s0 = edges[e * 3]; rl0 = edges[e * 3 + 1]; }
        if (c1 > 0) { int e = HALF + last1[n]; s1 = edges[e * 3]; rl1 = edges[e * 3 + 1]; }
        const float fc0 = (float)c0, fc1 = (float)c1;
        const float* self = rela_embed + (size_t)(2 * NUM_RELA) * D;
#pragma unroll
        for (int kb = 0; kb < 8; ++kb) {
            const int k = l + kb * 16;
            float ai = node_embed[(size_t)n * D + k] + self[k];
            float ao = (c0 > 0)
                ? fc0 * (node_embed[(size_t)s0 * D + k] + rela_embed[(size_t)rl0 * D + k]) : 0.f;
            float as = (c1 > 0)
                ? fc1 * (node_embed[(size_t)s1 * D + k] + rela_embed[(size_t)rl1 * D + k]) : 0.f;
            ldsA[r * LDS_STRIDE + k]         = ai;
            ldsA[r * LDS_STRIDE + D + k]     = ao;
            ldsA[r * LDS_STRIDE + 2 * D + k] = as;
        }
    }
    __syncthreads();

    // ---- each of 8 waves owns a 32x16 output block (two M tiles, shared B) ----
    const int wave    = t >> 5;
    const int lane    = t & 31;
    const int l16     = lane & 15;
    const int kOff    = (lane >> 4) << 1;   // 0 for lanes 0-15, 2 for lanes 16-31
    const int colBase = wave << 4;

    const float* aRow0 = &ldsA[l16 * LDS_STRIDE + kOff];
    const float* aRow1 = aRow0 + 16 * LDS_STRIDE;
    const float* bCol  = &WtP[(size_t)(colBase + l16) * 2];

    v8f acc0 = {};
    v8f acc1 = {};
#pragma unroll 8
    for (int kk = 0; kk < KTOT; kk += 4) {
        // B fragment: lane holds B[K=kk+kOff][N=colBase+l16], B[K=kk+kOff+1][N=...]
        v2f b = *(const v2f*)(bCol + (size_t)((kk + kOff) >> 1) * (2 * D));
        // A fragments: lane holds A[M=l16(+16)][kk+kOff], A[...][kk+kOff+1]
        v2f a0 = *(const v2f*)(aRow0 + kk);
        v2f a1 = *(const v2f*)(aRow1 + kk);
        acc0 = __builtin_amdgcn_wmma_f32_16x16x4_f32(
            false, a0, false, b, (short)0, acc0, false, false);
        acc1 = __builtin_amdgcn_wmma_f32_16x16x4_f32(
            false, a1, false, b, (short)0, acc1, false, false);
    }

    // ---- D layout: VGPR v -> M = v (+8 for upper half-wave), N = l16 ----
    const int mBase0 = rowBase + ((lane >> 4) << 3);
    const int mBase1 = mBase0 + 16;
#pragma unroll
    for (int v = 0; v < 8; ++v) {
        out[(size_t)(mBase0 + v) * D + colBase + l16] = acc0[v];
        out[(size_t)(mBase1 + v) * D + colBase + l16] = acc1[v];
    }
}

extern "C" void kernel_launch(void* const* d_in, const int* in_sizes, int n_in,
                              void* d_out, int out_size, void* d_ws, size_t ws_size,
                              hipStream_t stream) {
    (void)in_sizes; (void)n_in; (void)out_size; (void)ws_size;

    const float* node_embed = (const float*)d_in[0];
    const float* rela_embed = (const float*)d_in[1];
    const int*   edges      = (const int*)d_in[2];
    const float* W_o        = (const float*)d_in[3];
    const float* W_i        = (const float*)d_in[4];
    const float* W_s        = (const float*)d_in[5];
    const float* W_r        = (const float*)d_in[6];

    int*   last0 = (int*)d_ws;
    int*   cnt0  = last0 + N_NODES;
    int*   last1 = cnt0 + N_NODES;
    int*   cnt1  = last1 + N_NODES;
    float* WtP   = (float*)(cnt1 + N_NODES);   // 384*128 floats

    float* h_v = (float*)d_out;
    float* h_r = h_v + (size_t)N_NODES * D;

    init_ws_kernel<<<(4 * N_NODES + 255) / 256, 256, 0, stream>>>(last0, 4 * N_NODES);
    edge_scan_kernel<<<(N_EDGES + 255) / 256, 256, 0, stream>>>(edges, last0, cnt0, last1, cnt1);
    build_wt_kernel<<<(KTOT * D + 255) / 256, 256, 0, stream>>>(W_i, W_o, W_s, WtP);
    hr_kernel<<<(N_RELA_ROWS * D + 255) / 256, 256, 0, stream>>>(rela_embed, W_r, h_r);
    hv_wmma_kernel<<<N_NODES / ROWS_PER_WG, 256, 0, stream>>>(node_embed, rela_embed, edges,
                                                              last0, cnt0, last1, cnt1, WtP, h_v);
}